// iPUNetwork_4037269258634
// MI455X (gfx1250) — compile-verified
//
#include <hip/hip_runtime.h>

typedef __attribute__((ext_vector_type(2))) float v2f;
typedef __attribute__((ext_vector_type(8))) float v8f;

#define B_      2
#define N_      1024
#define C_      128
#define NS      48
#define GRIDG   7
#define NPTS    (B_ * N_)   // 2048

// output float offsets (concatenated tuple)
#define OUT_ORIS   0
#define OUT_NORS   6144
#define OUT_PTSUPS 12288
#define OUT_OFFSET 110592
#define OUT_PTSUP  116736

__device__ __forceinline__ void norm3(float* v) {
    float n = sqrtf(v[0]*v[0] + v[1]*v[1] + v[2]*v[2] + 1e-8f) + 1e-10f;
    v[0] /= n; v[1] /= n; v[2] /= n;
}
__device__ __forceinline__ void cross3(const float* a, const float* b, float* c) {
    c[0] = a[1]*b[2] - a[2]*b[1];
    c[1] = a[2]*b[0] - a[0]*b[2];
    c[2] = a[0]*b[1] - a[1]*b[0];
}

// ---------------- Kernel A: oris / nors / rotation matrix ----------------
__global__ __launch_bounds__(256) void kA(const float* __restrict__ feat,
                                          const float* __restrict__ W0,
                                          const float* __restrict__ b0,
                                          const float* __restrict__ W1,
                                          const float* __restrict__ b1,
                                          float* __restrict__ out,
                                          float* __restrict__ Mws) {
    const int w = threadIdx.x >> 5, lane = threadIdx.x & 31;
    const int pid = blockIdx.x * 8 + w;            // 0..2047
    const float4 f = *(const float4*)(feat + (size_t)pid * C_ + lane * 4);
    float s[6];
#pragma unroll
    for (int d = 0; d < 3; ++d) {
        const float4 w0 = *(const float4*)(W0 + d * C_ + lane * 4);
        const float4 w1 = *(const float4*)(W1 + d * C_ + lane * 4);
        s[d]     = f.x*w0.x + f.y*w0.y + f.z*w0.z + f.w*w0.w;
        s[3 + d] = f.x*w1.x + f.y*w1.y + f.z*w1.z + f.w*w1.w;
    }
#pragma unroll
    for (int off = 16; off >= 1; off >>= 1) {
#pragma unroll
        for (int i = 0; i < 6; ++i) s[i] += __shfl_xor(s[i], off);
    }
    float o[3], nr[3], r90[3], o0[3];
#pragma unroll
    for (int d = 0; d < 3; ++d) { o[d] = s[d] + b0[d]; nr[d] = s[3 + d] + b1[d]; }
    norm3(o); norm3(nr);
    cross3(o, nr, r90);  norm3(r90);
    cross3(r90, nr, o0); norm3(o0);
    if (lane == 0) {
#pragma unroll
        for (int d = 0; d < 3; ++d) {
            out[OUT_ORIS + pid * 3 + d] = o[d];
            out[OUT_NORS + pid * 3 + d] = nr[d];
            Mws[pid * 9 + d]     = o0[d];   // row0 = ori_0
            Mws[pid * 9 + 3 + d] = r90[d];  // row1 = ori_rot90
            Mws[pid * 9 + 6 + d] = nr[d];   // row2 = nors
        }
    }
}

// ---------------- Kernel B: ball query (first 48 in-radius, ascending) ----
__global__ __launch_bounds__(256) void kB(const float* __restrict__ xyz,
                                          int* __restrict__ gidxWs) {
    __shared__ float xs[N_], ys[N_], zs[N_], nrm[N_];
    const int b = blockIdx.x >> 7;                 // 128 blocks per batch
    const float* xb = xyz + b * 3 * N_;
    for (int j = threadIdx.x; j < N_; j += 256) {
        float x = xb[j], y = xb[N_ + j], z = xb[2 * N_ + j];
        xs[j] = x; ys[j] = y; zs[j] = z; nrm[j] = x*x + y*y + z*z;
    }
    __syncthreads();
    const int w = threadIdx.x >> 5, lane = threadIdx.x & 31;
    const int n = (blockIdx.x & 127) * 8 + w;
    const int pid = b * N_ + n;
    const float qx = xs[n], qy = ys[n], qz = zs[n];
    const float nq = nrm[n];
    int total = 0, firstj = -1;
    int* gp = gidxWs + (size_t)pid * NS;
    for (int c = 0; c < N_ / 32; ++c) {
        int j = c * 32 + lane;
        float d2 = -2.0f * (qx*xs[j] + qy*ys[j] + qz*zs[j]) + nq + nrm[j];
        bool in = (d2 <= 0.09f);
        unsigned mask = (unsigned)__ballot(in);
        if (in) {
            int pos = total + __popc(mask & ((1u << lane) - 1u));
            if (pos < NS) gp[pos] = j;
        }
        if (firstj < 0 && mask) firstj = c * 32 + (__ffs(mask) - 1);
        total += __popc(mask);
        if (total >= NS) break;                    // uniform across wave
    }
    for (int t = total + lane; t < NS; t += 32) gp[t] = firstj;
}

// ---------------- Kernel C: gather + WMMA GEMM + grid scatter + rotate ----
// One wave per point. A = group_feat (48x128 fp32, gathered from global),
// B = W_unet (128x21, zero-padded to 32 cols), pre-swizzled in LDS into
// WMMA fragment order so each K-step is one ds_load_b128 per lane.
// C = proj (48x21). V_WMMA_F32_16X16X4_F32, K stepped by 4.
__global__ __launch_bounds__(256) void kC(const float* __restrict__ xyz,
                                          const float* __restrict__ feat,
                                          const float* __restrict__ Wunet,
                                          const int*   __restrict__ indexSel,
                                          const float* __restrict__ Mws,
                                          const int*   __restrict__ gidxWs,
                                          float* __restrict__ out) {
    // Swizzled B panel: [kgroup(32)][half(2)][col lm(16)][4 floats]
    //   elem 0: B[4g+2h+0][lm]      elem 1: B[4g+2h+1][lm]
    //   elem 2: B[4g+2h+0][lm+16]   elem 3: B[4g+2h+1][lm+16]
    __shared__ float WcatSw[C_ * 32];
    __shared__ int   idxSh[16];
    __shared__ int   sIdx[8][NS];
    __shared__ int   sWin[8][NS];
    __shared__ float proj[8][NS * 24];             // cols 0..20 used

    const int tid = threadIdx.x;
    for (int i = tid; i < C_ * 32; i += 256) {
        const int g    = i >> 7;
        const int r    = i & 127;
        const int half = r >> 6;
        const int q    = r & 63;
        const int lm   = q >> 2;
        const int e    = q & 3;
        const int c    = g * 4 + 2 * half + (e & 1);
        const int nn   = lm + ((e >> 1) << 4);
        WcatSw[i] = (nn < 21) ? Wunet[c * 21 + nn] : 0.0f;   // col = k*3+d
    }
    if (tid < 16) idxSh[tid] = indexSel[tid];
    __syncthreads();

    const int w = tid >> 5, lane = tid & 31;
    const int pid = blockIdx.x * 8 + w;
    const int b = pid >> 10, n = pid & (N_ - 1);

    float M[9];
#pragma unroll
    for (int i = 0; i < 9; ++i) M[i] = Mws[pid * 9 + i];
    const float cx = xyz[b * 3 * N_ + n];
    const float cy = xyz[b * 3 * N_ + N_ + n];
    const float cz = xyz[b * 3 * N_ + 2 * N_ + n];
    const int* gix = gidxWs + (size_t)pid * NS;
    const float* featB = feat + (size_t)b * N_ * C_;

    // phase 0: per-sample flat cell index idx = 7*(cx+cy) + cz
    for (int s = lane; s < NS; s += 32) {
        int g = gix[s];
        __builtin_prefetch(featB + (size_t)g * C_, 0, 1);    // global_prefetch_b8
        float lx = xyz[b * 3 * N_ + g]           - cx;
        float ly = xyz[b * 3 * N_ + N_ + g]      - cy;
        float lz = xyz[b * 3 * N_ + 2 * N_ + g]  - cz;
        int cc[3];
#pragma unroll
        for (int d = 0; d < 3; ++d) {
            float lp = lx * M[d * 3 + 0] + ly * M[d * 3 + 1] + lz * M[d * 3 + 2];
            int ci = (int)rintf((lp + 0.3f) * 10.0f);        // (lp+R)/(2R)*(g-1)
            ci = min(GRIDG - 1, max(0, ci));
            cc[d] = ci;
        }
        sIdx[w][s] = (cc[0] + cc[1]) * GRIDG + cc[2];
    }
    __syncthreads();

    // winners (scatter .set semantics: last write to a cell wins)
    for (int s = lane; s < NS; s += 32) {
        int v = sIdx[w][s], win = 1;
        for (int s2 = s + 1; s2 < NS; ++s2) win &= (sIdx[w][s2] != v);
        sWin[w][s] = win;
    }

    // GEMM via fp32 WMMA. A frag (16x4): lanes 0-15 hold K={k0,k0+1},
    // lanes 16-31 hold K={k0+2,k0+3}. B frags from swizzled LDS: one
    // ds_load_b128 per lane per K-step yields both N-tile fragments.
    const int half = lane >> 4, lm = lane & 15;
    const float* bptr = WcatSw + half * 64 + lm * 4;
    for (int mt = 0; mt < 3; ++mt) {
        const float* arow = featB + (size_t)gix[mt * 16 + lm] * C_ + 2 * half;
        v8f acc0 = {}; v8f acc1 = {};
#pragma unroll 4
        for (int k0 = 0; k0 < C_; k0 += 4) {
            v2f a;  a.x = arow[k0]; a.y = arow[k0 + 1];
            const float4 bb = *(const float4*)(bptr + (k0 >> 2) * 128);
            v2f bf0; bf0.x = bb.x; bf0.y = bb.y;
            v2f bf1; bf1.x = bb.z; bf1.y = bb.w;
            acc0 = __builtin_amdgcn_wmma_f32_16x16x4_f32(
                       false, a, false, bf0, (short)0, acc0, false, false);
            acc1 = __builtin_amdgcn_wmma_f32_16x16x4_f32(
                       false, a, false, bf1, (short)0, acc1, false, false);
        }
        // C layout: vgpr v, lanes 0-15: M=v, N=lane; lanes 16-31: M=v+8
#pragma unroll
        for (int v = 0; v < 8; ++v) {
            int row = mt * 16 + v + 8 * half;
            proj[w][row * 24 + lm] = acc0[v];
            if (lm < 8) proj[w][row * 24 + 16 + lm] = acc1[v];
        }
    }
    __syncthreads();

    // phase 2: per-cell accumulation + rotation back to world space
    for (int cl = lane; cl < GRIDG * GRIDG; cl += 32) {
        float a0 = 0.f, a1 = 0.f, a2 = 0.f;
        for (int s = 0; s < NS; ++s) {
            int v = sIdx[w][s];
            int ij = v / GRIDG;
            if (ij != cl || !sWin[w][s]) continue;
            int k = v - ij * GRIDG;
            a0 += proj[w][s * 24 + k * 3 + 0];
            a1 += proj[w][s * 24 + k * 3 + 1];
            a2 += proj[w][s * 24 + k * 3 + 2];
        }
        float up0 = 0.1f * (float)(cl / GRIDG) - 0.3f + a0;
        float up1 = 0.1f * (float)(cl % GRIDG) - 0.3f + a1;
        float up2 = a2;
        float p0 = up0 * M[0] + up1 * M[3] + up2 * M[6] + cx;
        float p1 = up0 * M[1] + up1 * M[4] + up2 * M[7] + cy;
        float p2 = up0 * M[2] + up1 * M[5] + up2 * M[8] + cz;
        float* pu = out + OUT_PTSUP + ((size_t)pid * 49 + cl) * 3;
        pu[0] = p0; pu[1] = p1; pu[2] = p2;
        if (cl == 24) {
            float* po = out + OUT_OFFSET + (size_t)pid * 3;
            po[0] = p0; po[1] = p1; po[2] = p2;
        }
#pragma unroll
        for (int t = 0; t < 16; ++t) {
            if (idxSh[t] == cl) {
                float* pq = out + OUT_PTSUPS + ((size_t)pid * 16 + t) * 3;
                pq[0] = p0; pq[1] = p1; pq[2] = p2;
            }
        }
    }
}

extern "C" void kernel_launch(void* const* d_in, const int* in_sizes, int n_in,
                              void* d_out, int out_size, void* d_ws, size_t ws_size,
                              hipStream_t stream) {
    const float* xyz      = (const float*)d_in[0];
    const float* pts_feat = (const float*)d_in[1];
    const float* W0       = (const float*)d_in[2];
    const float* b0       = (const float*)d_in[3];
    const float* W1       = (const float*)d_in[4];
    const float* b1       = (const float*)d_in[5];
    const float* Wunet    = (const float*)d_in[6];
    const int*   indexSel = (const int*)d_in[7];
    float* out = (float*)d_out;

    float* Mws   = (float*)d_ws;                                   // 2048*9 f32
    int*   gidxW = (int*)((char*)d_ws + NPTS * 9 * sizeof(float)); // 2048*48 i32

    kA<<<NPTS / 8, 256, 0, stream>>>(pts_feat, W0, b0, W1, b1, out, Mws);
    kB<<<NPTS / 8, 256, 0, stream>>>(xyz, gidxW);
    kC<<<NPTS / 8, 256, 0, stream>>>(xyz, pts_feat, Wunet, indexSel, Mws, gidxW, out);
}